// Word2Vec_32341103739444
// MI455X (gfx1250) — compile-verified
//
#include <hip/hip_runtime.h>
#include <hip/hip_bf16.h>
#include <cstdint>

// ---------------------------------------------------------------------------
// Problem constants (match reference)
// ---------------------------------------------------------------------------
constexpr int ORIG = 32000;
constexpr int NEWR = 31199;
constexpr int FACT = 1024;
constexpr int D    = 1024;
constexpr int Bsz  = 4096;
constexpr int Kneg = 10;

constexpr int CW        = 256;            // vocab chunk width (fused softmax)
constexpr int KB        = 16;             // K rows per async-staged panel block
constexpr int PHASES    = FACT / KB;      // 64
constexpr int SB_STRIDE = CW + 16;        // 272: rows 2 apart -> disjoint banks
constexpr int SS_STRIDE = CW + 1;         // 257: odd stride, conflict-free col reads
static_assert(ORIG % CW == 0, "chunking");

// ---------------------------------------------------------------------------
// CDNA5 WMMA: V_WMMA_F32_16X16X4_F32  (D = A[16x4] * B[4x16] + C[16x16])
// ---------------------------------------------------------------------------
typedef __attribute__((ext_vector_type(2))) float v2f;
typedef __attribute__((ext_vector_type(8))) float v8f;
typedef __attribute__((ext_vector_type(4))) int   v4i;

static __device__ __forceinline__ v8f wmma_f32(v2f a, v2f b, v8f c) {
  return __builtin_amdgcn_wmma_f32_16x16x4_f32(false, a, false, b,
                                               (short)0, c, false, false);
}

// ---------------------------------------------------------------------------
// CDNA5 async global->LDS copy (ASYNCcnt path).
// Probe decoded the builtin's param 1 as `v4i addrspace(1)*`
// (hipcc prints addrspace(1) as `__device__`), so pass typed v4i pointers;
// these also convert implicitly if a param is `void*` in the same AS.
// ---------------------------------------------------------------------------
typedef __attribute__((address_space(1))) v4i* gv4i_p;
typedef __attribute__((address_space(3))) v4i* lv4i_p;

static __device__ __forceinline__ void asyncLoad16B(const float* g, float* l) {
  gv4i_p gp = (gv4i_p)g;
  lv4i_p lp = (lv4i_p)l;
#if __has_builtin(__builtin_amdgcn_global_load_async_to_lds_b128)
  __builtin_amdgcn_global_load_async_to_lds_b128(gp, lp, 0, 0);
#else
  unsigned ldsAddr = (unsigned)(size_t)lp;
  asm volatile("global_load_async_to_lds_b128 %0, %1, off"
               :: "v"(ldsAddr), "v"(gp) : "memory");
#endif
}

static __device__ __forceinline__ void waitAsync0() {
#if __has_builtin(__builtin_amdgcn_s_wait_asynccnt)
  __builtin_amdgcn_s_wait_asynccnt(0);
#else
  asm volatile("s_wait_asynccnt 0x0" ::: "memory");
#endif
}

// ---------------------------------------------------------------------------
// Kernel 0: zero counters + row flags
// ---------------------------------------------------------------------------
__global__ void initKernel(int* counters, int* flagsW, int* flagsV) {
  int i = blockIdx.x * blockDim.x + threadIdx.x;
  if (i < 2) counters[i] = 0;
  if (i < NEWR) { flagsW[i] = 0; flagsV[i] = 0; }
}

// ---------------------------------------------------------------------------
// Kernel 1: mark referenced "new" rows
// ---------------------------------------------------------------------------
__global__ void markKernel(const int* __restrict__ in_ids,
                           const int* __restrict__ pos_ids,
                           const int* __restrict__ neg_ids,
                           int* flagsW, int* flagsV) {
  int i = blockIdx.x * blockDim.x + threadIdx.x;
  if (i < Bsz) {
    int v = in_ids[i];
    if (v >= ORIG) flagsW[v - ORIG] = 1;
  } else if (i < 2 * Bsz) {
    int v = pos_ids[i - Bsz];
    if (v >= ORIG) flagsV[v - ORIG] = 1;
  } else if (i < 2 * Bsz + Bsz * Kneg) {
    int v = neg_ids[i - 2 * Bsz];
    if (v >= ORIG) flagsV[v - ORIG] = 1;
  }
}

// ---------------------------------------------------------------------------
// Kernel 2: compact flagged rows (per-row results are slot-independent, so
// atomic ordering does not affect the final output bits)
// ---------------------------------------------------------------------------
__global__ void compactKernel(const int* __restrict__ flagsW,
                              const int* __restrict__ flagsV,
                              int* listW, int* listV, int* counters) {
  int i = blockIdx.x * blockDim.x + threadIdx.x;
  if (i >= NEWR) return;
  if (flagsW[i]) listW[atomicAdd(&counters[0], 1)] = i;
  if (flagsV[i]) listV[atomicAdd(&counters[1], 1)] = i;
}

// ---------------------------------------------------------------------------
// Kernel 3: tab[r] = softmax(A1[r] @ A2) @ M for a tile of 16 needed rows.
// Block = 128 threads = 4 waves. Per CW-column chunk:
//   GEMM1: Z[16,CW] = A1tile @ A2[:,chunk]   (B panel async-staged into LDS,
//          double buffered in KB-row blocks; 4 indep WMMA chains per wave)
//   online softmax (parallel row max / exp / sum)
//   GEMM2: O[16,1024] += S[16,CW] @ M[chunk,:] (kk-outer, 16 indep chains)
// ---------------------------------------------------------------------------
__global__ __launch_bounds__(128) void rowTileKernel(
    const float* __restrict__ A1, const float* __restrict__ A2,
    const float* __restrict__ M, const int* __restrict__ list,
    const int* __restrict__ count, float* __restrict__ tab) {
  __shared__ float sB[2][KB][SB_STRIDE];   // async-staged A2 panel (34.8 KB)
  __shared__ float sS[16][SS_STRIDE];      // exp'd logit chunk (16.4 KB)
  __shared__ float sP[16][8];              // per-row partials
  __shared__ float sMax[16], sSum[16], sScale[16];
  __shared__ int   sRow[16];

  const int cnt = *count;
  if (blockIdx.x * 16 >= cnt) return;      // uniform early-exit

  const int tid    = threadIdx.x;
  const int wave   = tid >> 5;             // 0..3
  const int lane   = tid & 31;
  const int laneM  = lane & 15;
  const int laneHi = lane >> 4;

  const int prow = tid >> 3;               // staging/stats: 8 threads per row
  const int pcol = (tid & 7) * 32;

  if (tid < 16) {
    int idx = blockIdx.x * 16 + tid;
    sRow[tid] = (idx < cnt) ? list[idx] : -1;
    sMax[tid] = -3.4e38f;
    sSum[tid] = 0.0f;
  }
  __syncthreads();

  int rA = sRow[laneM] < 0 ? 0 : sRow[laneM];
  const float* aRow = A1 + (size_t)rA * FACT;

  v8f O[16] = {};                          // wave owns cols [wave*256,+256)

  for (int c0 = 0; c0 < ORIG; c0 += CW) {
    // token prefetch of next chunk's GEMM2 panel rows
    if (c0 + CW < ORIG) {
      __builtin_prefetch(M + (size_t)(c0 + CW + tid) * D, 0, 1);
      __builtin_prefetch(M + (size_t)(c0 + CW + 128 + tid) * D, 0, 1);
    }

    // ---- GEMM1 with async double-buffered B panel -------------------------
    v8f z[4] = {};
    {
      // prologue: stage block 0
      {
        const float* src = A2 + (size_t)prow * ORIG + c0 + pcol;
        float* dst = &sB[0][prow][pcol];
        #pragma unroll
        for (int j = 0; j < 8; ++j) asyncLoad16B(src + j * 4, dst + j * 4);
      }
      waitAsync0();
      __syncthreads();

      for (int kb = 0; kb < PHASES; ++kb) {
        const int buf = kb & 1;
        if (kb + 1 < PHASES) {             // stage next block into other buffer
          const float* src =
              A2 + (size_t)((kb + 1) * KB + prow) * ORIG + c0 + pcol;
          float* dst = &sB[buf ^ 1][prow][pcol];
          #pragma unroll
          for (int j = 0; j < 8; ++j) asyncLoad16B(src + j * 4, dst + j * 4);
        }
        // compute on current block: KB=16 -> 4 K-steps, 4 subtiles each
        const int kbase = kb * KB;
        #pragma unroll
        for (int kl = 0; kl < KB; kl += 4) {
          v2f a = *(const v2f*)(aRow + kbase + kl + laneHi * 2);
          const float* b0 = &sB[buf][kl + laneHi * 2][0];
          v2f b[4];
          #pragma unroll
          for (int sub = 0; sub < 4; ++sub) {
            const int n = wave * 64 + sub * 16 + laneM;
            b[sub].x = b0[n];
            b[sub].y = b0[SB_STRIDE + n];
          }
          #pragma unroll
          for (int sub = 0; sub < 4; ++sub) z[sub] = wmma_f32(a, b[sub], z[sub]);
        }
        if (kb + 1 < PHASES) waitAsync0();
        __syncthreads();
      }
    }

    // spill Z to LDS
    #pragma unroll
    for (int sub = 0; sub < 4; ++sub) {
      const int cl = wave * 64 + sub * 16 + laneM;
      #pragma unroll
      for (int i = 0; i < 8; ++i) sS[i + laneHi * 8][cl] = z[sub][i];
    }
    __syncthreads();

    // ---- online softmax: parallel max -------------------------------------
    {
      float m = -3.4e38f;
      for (int j = 0; j < 32; ++j) m = fmaxf(m, sS[prow][pcol + j]);
      sP[prow][tid & 7] = m;
    }
    __syncthreads();
    if (tid < 16) {
      float mNew = sMax[tid];
      #pragma unroll
      for (int j = 0; j < 8; ++j) mNew = fmaxf(mNew, sP[tid][j]);
      sScale[tid] = __expf(sMax[tid] - mNew);
      sMax[tid] = mNew;
    }
    __syncthreads();
    // ---- exp + partial sums ----------------------------------------------
    {
      const float mNew = sMax[prow];
      float s = 0.0f;
      for (int j = 0; j < 32; ++j) {
        float e = __expf(sS[prow][pcol + j] - mNew);
        sS[prow][pcol + j] = e;
        s += e;
      }
      sP[prow][tid & 7] = s;
    }
    __syncthreads();
    if (tid < 16) {
      float l = sSum[tid] * sScale[tid];
      #pragma unroll
      for (int j = 0; j < 8; ++j) l += sP[tid][j];
      sSum[tid] = l;
    }
    __syncthreads();

    // ---- rescale accumulators by exp(mOld - mNew) -------------------------
    float scRow[8];
    #pragma unroll
    for (int j = 0; j < 8; ++j) scRow[j] = sScale[j + laneHi * 8];
    #pragma unroll
    for (int t = 0; t < 16; ++t)
      #pragma unroll
      for (int j = 0; j < 8; ++j) O[t][j] *= scRow[j];

    // ---- GEMM2: kk-outer, 16 independent chains ---------------------------
    for (int kk = 0; kk < CW; kk += 4) {
      v2f a = *(const v2f*)&sS[laneM][kk + laneHi * 2];
      const float* m0 =
          M + (size_t)(c0 + kk + laneHi * 2) * D + wave * 256 + laneM;
      v2f b[16];
      #pragma unroll
      for (int t = 0; t < 16; ++t) {
        b[t].x = m0[t * 16];
        b[t].y = m0[D + t * 16];
      }
      #pragma unroll
      for (int t = 0; t < 16; ++t) O[t] = wmma_f32(a, b[t], O[t]);
    }
    __syncthreads();   // sS / sB reused next chunk
  }

  // ---- normalize and scatter rows ----------------------------------------
  float invL[8];
  #pragma unroll
  for (int j = 0; j < 8; ++j) invL[j] = 1.0f / sSum[j + laneHi * 8];
  #pragma unroll
  for (int t = 0; t < 16; ++t) {
    const int n = wave * 256 + t * 16 + laneM;
    #pragma unroll
    for (int j = 0; j < 8; ++j) {
      const int r = sRow[j + laneHi * 8];
      if (r >= 0) tab[(size_t)r * D + n] = O[t][j] * invL[j];
    }
  }
}

// ---------------------------------------------------------------------------
// Kernel 4: per-sample loss.  pos = e·p ;  negTot = e·(Σk negk)
// ---------------------------------------------------------------------------
__global__ __launch_bounds__(256) void lossKernel(
    const float* __restrict__ Worig, const float* __restrict__ Vorig,
    const float* __restrict__ tabW, const float* __restrict__ tabV,
    const int* __restrict__ in_ids, const int* __restrict__ pos_ids,
    const int* __restrict__ neg_ids, float* __restrict__ lossArr) {
  const int b = blockIdx.x;
  const int tid = threadIdx.x;

  const int ii = in_ids[b];
  const float* e = (ii < ORIG) ? (Worig + (size_t)ii * D)
                               : (tabW + (size_t)(ii - ORIG) * D);
  const int pi = pos_ids[b];
  const float* p = (pi < ORIG) ? (Vorig + (size_t)pi * D)
                               : (tabV + (size_t)(pi - ORIG) * D);
  const float* np[Kneg];
  #pragma unroll
  for (int k = 0; k < Kneg; ++k) {
    const int ni = neg_ids[b * Kneg + k];
    np[k] = (ni < ORIG) ? (Vorig + (size_t)ni * D)
                        : (tabV + (size_t)(ni - ORIG) * D);
  }

  float dp = 0.0f, dn = 0.0f;
  for (int d = tid; d < D; d += 256) {
    const float ev = e[d];
    dp += p[d] * ev;
    float s = 0.0f;
    #pragma unroll
    for (int k = 0; k < Kneg; ++k) s += np[k][d];
    dn += s * ev;
  }

  __shared__ float r1[256], r2[256];
  r1[tid] = dp; r2[tid] = dn;
  __syncthreads();
  for (int st = 128; st > 0; st >>= 1) {
    if (tid < st) { r1[tid] += r1[tid + st]; r2[tid] += r2[tid + st]; }
    __syncthreads();
  }
  if (tid == 0) {
    const float pos = r1[0], neg = r2[0];
    float x = -pos;
    float lp = fmaxf(x, 0.0f) + log1pf(__expf(-fabsf(x)));
    float ln = fmaxf(neg, 0.0f) + log1pf(__expf(-fabsf(neg)));
    lossArr[b] = lp + ln;
  }
}

// ---------------------------------------------------------------------------
// Kernel 5: fixed-order mean reduction (bitwise deterministic)
// ---------------------------------------------------------------------------
__global__ __launch_bounds__(256) void reduceKernel(
    const float* __restrict__ lossArr, float* __restrict__ out) {
  const int tid = threadIdx.x;
  float s = 0.0f;
  for (int i = tid; i < Bsz; i += 256) s += lossArr[i];
  __shared__ float r[256];
  r[tid] = s;
  __syncthreads();
  for (int st = 128; st > 0; st >>= 1) {
    if (tid < st) r[tid] += r[tid + st];
    __syncthreads();
  }
  if (tid == 0) out[0] = r[0] / (float)Bsz;
}

// ---------------------------------------------------------------------------
// kernel_launch — workspace ~256.5 MB:
//   [counters][flagsW][flagsV][listW][listV][lossArr][tabW NEWR*D][tabV NEWR*D]
// ---------------------------------------------------------------------------
extern "C" void kernel_launch(void* const* d_in, const int* in_sizes, int n_in,
                              void* d_out, int out_size, void* d_ws,
                              size_t ws_size, hipStream_t stream) {
  (void)in_sizes; (void)n_in; (void)out_size; (void)ws_size;

  const float* W_original = (const float*)d_in[0];
  const float* V_original = (const float*)d_in[1];
  const float* A_W_1      = (const float*)d_in[2];
  const float* A_W_2      = (const float*)d_in[3];
  const float* A_V_1      = (const float*)d_in[4];
  const float* A_V_2      = (const float*)d_in[5];
  const int*   in_ids     = (const int*)d_in[6];
  const int*   pos_ids    = (const int*)d_in[7];
  const int*   neg_ids    = (const int*)d_in[8];
  float*       out        = (float*)d_out;

  char* ws = (char*)d_ws;
  int*   counters = (int*)ws;
  int*   flagsW   = (int*)(ws + 256);
  int*   flagsV   = flagsW + NEWR;
  int*   listW    = flagsV + NEWR;
  int*   listV    = listW + NEWR;
  float* lossArr  = (float*)(listV + NEWR);
  size_t off = 256 + 4ull * sizeof(int) * NEWR + sizeof(float) * Bsz;
  off = (off + 255) & ~(size_t)255;
  float* tabW = (float*)(ws + off);
  float* tabV = tabW + (size_t)NEWR * D;

  initKernel<<<(NEWR + 255) / 256, 256, 0, stream>>>(counters, flagsW, flagsV);
  markKernel<<<(2 * Bsz + Bsz * Kneg + 255) / 256, 256, 0, stream>>>(
      in_ids, pos_ids, neg_ids, flagsW, flagsV);
  compactKernel<<<(NEWR + 255) / 256, 256, 0, stream>>>(flagsW, flagsV, listW,
                                                        listV, counters);
  rowTileKernel<<<(Bsz + 15) / 16, 128, 0, stream>>>(A_W_1, A_W_2, W_original,
                                                     listW, &counters[0], tabW);
  rowTileKernel<<<(NEWR + 15) / 16, 128, 0, stream>>>(A_V_1, A_V_2, V_original,
                                                      listV, &counters[1], tabV);
  lossKernel<<<Bsz, 256, 0, stream>>>(W_original, V_original, tabW, tabV,
                                      in_ids, pos_ids, neg_ids, lossArr);
  reduceKernel<<<1, 256, 0, stream>>>(lossArr, out);
}